// GraphSAGE_52690658787597
// MI455X (gfx1250) — compile-verified
//
#include <hip/hip_runtime.h>
#include <hip/hip_bf16.h>

#define N_NODES 50000
#define N_EDGES 800000
#define D_FEAT  128
#define D_EDGE  64
#define EMBED   128
#define WPB     8   // waves per block

typedef __attribute__((ext_vector_type(2))) float v2f;
typedef __attribute__((ext_vector_type(8))) float v8f;

// ---------------------------------------------------------------------------
// Degree accumulation
// ---------------------------------------------------------------------------
__global__ void zero_deg_kernel(float* __restrict__ deg) {
    int i = blockIdx.x * blockDim.x + threadIdx.x;
    if (i < N_NODES) deg[i] = 0.0f;
}

__global__ void degree_kernel(const int* __restrict__ recv, float* __restrict__ deg) {
    int e = blockIdx.x * blockDim.x + threadIdx.x;
    if (e < N_EDGES) atomicAdd(&deg[recv[e]], 1.0f);
}

// ---------------------------------------------------------------------------
// Node GEMM: hW = X @ W + b.  X:[N,128], W:[128,256].
// Cols [0,128)  -> h_self -> written straight into d_out (nodes accumulator)
// Cols [128,256)-> h      -> workspace (gathered later per edge)
// One wave per 16x16 output tile; 50000 tiles exactly (3125 row x 16 col).
// ---------------------------------------------------------------------------
__global__ void __launch_bounds__(256)
node_gemm_kernel(const float* __restrict__ X,
                 const float* __restrict__ W,
                 const float* __restrict__ bias,
                 float* __restrict__ out_self,   // d_out [N,128]
                 float* __restrict__ h)          // ws    [N,128]
{
    const int wave = (blockIdx.x * blockDim.x + threadIdx.x) >> 5;
    const int lane = threadIdx.x & 31;
    const int tm = wave >> 4;          // 0..3124
    const int tn = wave & 15;          // 0..15
    const int m0 = tm * 16, n0 = tn * 16;
    const int lc = lane & 15;
    const int half = lane >> 4;
    const int khalf = half * 2;        // K offset 0 or 2 within a 4-wide K step
    const int row = m0 + lc;

    v8f acc = {};
#pragma unroll
    for (int k0 = 0; k0 < D_FEAT; k0 += 4) {
        const int ka = k0 + khalf;
        v2f a = *(const v2f*)(X + (size_t)row * D_FEAT + ka);     // A[M=row][ka, ka+1]
        v2f b;
        b.x = W[(size_t)ka       * 256 + n0 + lc];                // B[K=ka  ][N]
        b.y = W[(size_t)(ka + 1) * 256 + n0 + lc];                // B[K=ka+1][N]
        acc = __builtin_amdgcn_wmma_f32_16x16x4_f32(
                  false, a, false, b, (short)0, acc, false, false);
    }

    const float bv = bias[n0 + lc];
    const int mbase = m0 + half * 8;
    const int col = n0 + lc;
#pragma unroll
    for (int r = 0; r < 8; ++r) {
        const float v = acc[r] + bv;
        const int mr = mbase + r;
        if (col < EMBED) out_self[(size_t)mr * EMBED + col] = v;
        else             h[(size_t)mr * EMBED + (col - EMBED)] = v;
    }
}

// ---------------------------------------------------------------------------
// Edge kernel: per 16-edge tile (one wave):
//   mlp  = EF[e] @ We + be              (WMMA, We staged in LDS: 32 KB)
//   val  = (mlp + h[senders[e]]) * max(deg[receivers[e]], 1)
//   atomicAdd(out[receivers[e]], val)   (64B-coalesced f32 atomics)
// 50000 tiles exactly; 8 waves/block -> 6250 blocks, no stragglers.
// ---------------------------------------------------------------------------
__global__ void __launch_bounds__(256)
edge_kernel(const float* __restrict__ EF,        // [E,64]
            const float* __restrict__ We,        // [64,128]
            const float* __restrict__ Web,       // [128]
            const int*   __restrict__ senders,
            const int*   __restrict__ receivers,
            const float* __restrict__ h,         // ws [N,128]
            const float* __restrict__ deg,       // ws [N]
            float* __restrict__ out)             // d_out [N,128]
{
    __shared__ float sWe[D_EDGE * EMBED];        // 32 KB
    __shared__ float sBias[EMBED];
    __shared__ int   sSend[WPB][16];
    __shared__ int   sRecv[WPB][16];
    __shared__ float sScale[WPB][16];

    for (int i = threadIdx.x; i < D_EDGE * EMBED; i += blockDim.x) sWe[i] = We[i];
    for (int i = threadIdx.x; i < EMBED; i += blockDim.x) sBias[i] = Web[i];
    __syncthreads();

    const int wib  = threadIdx.x >> 5;
    const int lane = threadIdx.x & 31;
    const int tile = blockIdx.x * WPB + wib;     // 0..49999, exact
    const int e0   = tile * 16;
    const int lc   = lane & 15;
    const int half = lane >> 4;
    const int khalf = half * 2;

    // per-edge metadata (cross-lane via LDS; same-wave LDS ops are in-order)
    if (lane < 16) {
        const int s = senders[e0 + lane];
        const int r = receivers[e0 + lane];
        const float d = deg[r];
        sSend[wib][lane]  = s;
        sRecv[wib][lane]  = r;
        sScale[wib][lane] = d < 1.0f ? 1.0f : d;
    }

    // Preload A fragments: 16 K-steps x (2 floats/lane) = 32 VGPRs
    const int erow = e0 + lc;
    v2f afrag[16];
#pragma unroll
    for (int kk = 0; kk < 16; ++kk)
        afrag[kk] = *(const v2f*)(EF + (size_t)erow * D_EDGE + kk * 4 + khalf);

#pragma unroll
    for (int nt = 0; nt < 8; ++nt) {
        const int n0 = nt * 16;
        v8f acc = {};
#pragma unroll
        for (int kk = 0; kk < 16; ++kk) {
            const int ka = kk * 4 + khalf;
            v2f b;
            b.x = sWe[ka       * EMBED + n0 + lc];
            b.y = sWe[(ka + 1) * EMBED + n0 + lc];
            acc = __builtin_amdgcn_wmma_f32_16x16x4_f32(
                      false, afrag[kk], false, b, (short)0, acc, false, false);
        }
        const float bv = sBias[n0 + lc];
#pragma unroll
        for (int r = 0; r < 8; ++r) {
            const int m   = half * 8 + r;               // edge slot in tile
            const int s   = sSend[wib][m];
            const int rcv = sRecv[wib][m];
            const float sc = sScale[wib][m];
            const float hv = h[(size_t)s * EMBED + n0 + lc];    // coalesced 64B gather
            const float v  = (acc[r] + bv + hv) * sc;
            atomicAdd(&out[(size_t)rcv * EMBED + n0 + lc], v);  // coalesced 64B atomic
        }
    }
}

// ---------------------------------------------------------------------------
extern "C" void kernel_launch(void* const* d_in, const int* in_sizes, int n_in,
                              void* d_out, int out_size, void* d_ws, size_t ws_size,
                              hipStream_t stream) {
    const float* node_features = (const float*)d_in[0];   // [N,128]
    const int*   senders       = (const int*)  d_in[1];   // [E]
    const int*   receivers     = (const int*)  d_in[2];   // [E]
    const float* edge_features = (const float*)d_in[3];   // [E,64]
    const float* W_kernel      = (const float*)d_in[4];   // [128,256]
    const float* W_bias        = (const float*)d_in[5];   // [256]
    const float* We_kernel     = (const float*)d_in[6];   // [64,128]
    const float* We_bias       = (const float*)d_in[7];   // [128]
    float* out = (float*)d_out;                           // [N,128]

    float* ws_h   = (float*)d_ws;                         // [N,128] = 25.6 MB
    float* ws_deg = ws_h + (size_t)N_NODES * EMBED;       // [N]

    zero_deg_kernel<<<(N_NODES + 255) / 256, 256, 0, stream>>>(ws_deg);
    degree_kernel<<<(N_EDGES + 255) / 256, 256, 0, stream>>>(receivers, ws_deg);

    // 50000 wave-tiles / 8 waves per block
    node_gemm_kernel<<<6250, 256, 0, stream>>>(node_features, W_kernel, W_bias,
                                               out, ws_h);
    edge_kernel<<<6250, 256, 0, stream>>>(edge_features, We_kernel, We_bias,
                                          senders, receivers, ws_h, ws_deg, out);
}